// SPIntraAttModuleV4_33346126086743
// MI455X (gfx1250) — compile-verified
//
#include <hip/hip_runtime.h>
#include <math.h>

// Problem constants (fixed shapes from the reference)
#define NPIX   147456      // 384*384
#define CDIM   64
#define QKD    32
#define NSP    576
#define TOPK_  32
#define NSAMP  30
#define EPSLN  1e-5f

typedef float v2f __attribute__((ext_vector_type(2)));
typedef float v8f __attribute__((ext_vector_type(8)));

// ---------------------------------------------------------------------------
// Kernel 1: per-pixel LayerNorm over C=64, then [128x64]x[64xN] f32 WMMA GEMM
// producing q (32 rows), k (32 rows), v (64 rows). v is also written to d_out.
// One wave = 32 pixels. Block = 2 waves. LDS: W (128x65) + 2 x tiles (64x33).
// ---------------------------------------------------------------------------
__global__ __launch_bounds__(64) void k_ln_qkv(
    const float* __restrict__ x,  const float* __restrict__ Wq,
    const float* __restrict__ Wk, const float* __restrict__ Wv,
    const float* __restrict__ gamma, const float* __restrict__ beta,
    float* __restrict__ qo, float* __restrict__ ko, float* __restrict__ vo,
    float* __restrict__ out)
{
    __shared__ float Wl[128 * 65];
    __shared__ float xs[2][64 * 33];
    const int tid  = threadIdx.x;
    const int lane = tid & 31;
    const int w    = tid >> 5;

    // Stage stacked W = [Wq; Wk; Wv] into LDS, row-padded to 65 floats.
    for (int i = tid; i < 128 * 64; i += 64) {
        int o = i >> 6, c = i & 63;
        float val;
        if (o < 32)       val = Wq[o * 64 + c];
        else if (o < 64)  val = Wk[(o - 32) * 64 + c];
        else              val = Wv[(o - 64) * 64 + c];
        Wl[o * 65 + c] = val;
    }

    const int n0 = blockIdx.x * 64 + w * 32;   // base pixel for this wave
    float* X = xs[w];
    for (int c = 0; c < 64; ++c)
        X[c * 33 + lane] = x[(size_t)c * NPIX + n0 + lane];
    __syncthreads();

    // Two-pass layernorm (lane owns pixel n0+lane).
    float mu = 0.f;
    #pragma unroll
    for (int c = 0; c < 64; ++c) mu += X[c * 33 + lane];
    mu *= (1.f / 64.f);
    float var = 0.f;
    #pragma unroll
    for (int c = 0; c < 64; ++c) { float d = X[c * 33 + lane] - mu; var += d * d; }
    var *= (1.f / 64.f);
    float rstd = rsqrtf(var + EPSLN);
    for (int c = 0; c < 64; ++c)
        X[c * 33 + lane] = (X[c * 33 + lane] - mu) * rstd * gamma[c] + beta[c];
    __syncthreads();

    // WMMA: D[128 x 32] = W[128 x 64] * Xn[64 x 32], f32 16x16x4 tiles.
    const int mrow = lane & 15;
    const int kh   = (lane < 16) ? 0 : 2;  // A/B VGPR0 holds K=0 | K=2 halves
    for (int sub = 0; sub < 2; ++sub) {
        const int p0 = sub * 16;
        for (int blk = 0; blk < 8; ++blk) {
            v8f acc = {0.f, 0.f, 0.f, 0.f, 0.f, 0.f, 0.f, 0.f};
            #pragma unroll
            for (int kk = 0; kk < 16; ++kk) {
                const int c0 = kk * 4 + kh;
                v2f a, b;
                a.x = Wl[(blk * 16 + mrow) * 65 + c0];
                a.y = Wl[(blk * 16 + mrow) * 65 + c0 + 1];
                b.x = X[c0 * 33 + p0 + mrow];
                b.y = X[(c0 + 1) * 33 + p0 + mrow];
                acc = __builtin_amdgcn_wmma_f32_16x16x4_f32(
                        false, a, false, b, (short)0, acc, false, false);
            }
            const int rbase = blk * 16 + ((lane < 16) ? 0 : 8);
            const int col   = n0 + p0 + mrow;
            #pragma unroll
            for (int r = 0; r < 8; ++r) {
                int row = rbase + r;
                float vv = acc[r];
                if (row < 32)       qo[(size_t)row * NPIX + col] = vv;
                else if (row < 64)  ko[(size_t)(row - 32) * NPIX + col] = vv;
                else {
                    vo [(size_t)(row - 64) * NPIX + col] = vv;
                    out[(size_t)(row - 64) * NPIX + col] = vv;   // res starts as v
                }
            }
        }
    }
}

// ---------------------------------------------------------------------------
// Kernel 2: labels[n] = argmax_k affinity[k][n] (first index on ties).
// ---------------------------------------------------------------------------
__global__ __launch_bounds__(256) void k_labels(const float* __restrict__ aff,
                                                int* __restrict__ labels)
{
    const int n = blockIdx.x * 256 + threadIdx.x;
    float best = aff[n];
    int   bl   = 0;
    for (int k = 1; k < NSP; ++k) {
        int kp = (k + 4 < NSP) ? (k + 4) : (NSP - 1);
        __builtin_prefetch(&aff[(size_t)kp * NPIX + n], 0, 1);
        float v = aff[(size_t)k * NPIX + n];
        if (v > best) { best = v; bl = k; }
    }
    labels[n] = bl;
}

// ---------------------------------------------------------------------------
// Kernel 3: per-row top-32 (value desc, index asc tie-break). One block/row.
// ---------------------------------------------------------------------------
__global__ __launch_bounds__(128) void k_topk(const float* __restrict__ aff,
                                              float* __restrict__ topv,
                                              int* __restrict__ topi)
{
    __shared__ float tv[128 * 32];
    __shared__ int   tix[128 * 32];
    __shared__ float rv[128];
    __shared__ int   rpx[128];
    __shared__ int   rslot[128];
    const int   tid = threadIdx.x;
    const int   r   = blockIdx.x;
    const float NEG = -3.4e38f;

    float* mv = &tv[tid * 32];
    int*   mi = &tix[tid * 32];
    for (int j = 0; j < 32; ++j) { mv[j] = NEG; mi[j] = 0x7fffffff; }

    float curmin = NEG; int minslot = 0;
    const float* row = aff + (size_t)r * NPIX;
    for (int i = tid; i < NPIX; i += 128) {
        __builtin_prefetch(&row[i + 512], 0, 1);
        float v = row[i];
        if (v > curmin) {
            mv[minslot] = v; mi[minslot] = i;
            curmin = mv[0]; minslot = 0;
            for (int j = 1; j < 32; ++j)
                if (mv[j] < curmin) { curmin = mv[j]; minslot = j; }
        }
    }
    __syncthreads();

    for (int sel = 0; sel < 32; ++sel) {
        float bv = mv[0]; int bslot = 0; int bpix = mi[0];
        for (int j = 1; j < 32; ++j) {
            float v = mv[j];
            if (v > bv || (v == bv && mi[j] < bpix)) { bv = v; bslot = j; bpix = mi[j]; }
        }
        rv[tid] = bv; rpx[tid] = bpix; rslot[tid] = tid * 32 + bslot;
        __syncthreads();
        for (int off = 64; off > 0; off >>= 1) {
            if (tid < off) {
                float v2 = rv[tid + off];
                if (v2 > rv[tid] || (v2 == rv[tid] && rpx[tid + off] < rpx[tid])) {
                    rv[tid] = v2; rpx[tid] = rpx[tid + off]; rslot[tid] = rslot[tid + off];
                }
            }
            __syncthreads();
        }
        if (tid == 0) {
            topv[r * 32 + sel] = rv[0];
            topi[r * 32 + sel] = rpx[0];
            tv [rslot[0]] = NEG;
            tix[rslot[0]] = 0x7fffffff;
        }
        __syncthreads();
    }
}

// ---------------------------------------------------------------------------
// threefry2x32 matching JAX's default PRNG (key = (0, seed)).
// ---------------------------------------------------------------------------
__device__ __forceinline__ unsigned rotl_(unsigned x, int d) {
    return (x << d) | (x >> (32 - d));
}
__device__ __forceinline__ void threefry2x32(unsigned k0, unsigned k1,
                                             unsigned x0, unsigned x1,
                                             unsigned& o0, unsigned& o1)
{
    unsigned ks2 = k0 ^ k1 ^ 0x1BD11BDAu;
    x0 += k0; x1 += k1;
    const int RA[4] = {13, 15, 26, 6};
    const int RB[4] = {17, 29, 16, 24};
    #pragma unroll
    for (int i = 0; i < 4; ++i) { x0 += x1; x1 = rotl_(x1, RA[i]); x1 ^= x0; }
    x0 += k1;  x1 += ks2 + 1u;
    #pragma unroll
    for (int i = 0; i < 4; ++i) { x0 += x1; x1 = rotl_(x1, RB[i]); x1 ^= x0; }
    x0 += ks2; x1 += k0 + 2u;
    #pragma unroll
    for (int i = 0; i < 4; ++i) { x0 += x1; x1 = rotl_(x1, RA[i]); x1 ^= x0; }
    x0 += k0;  x1 += k1 + 3u;
    #pragma unroll
    for (int i = 0; i < 4; ++i) { x0 += x1; x1 = rotl_(x1, RB[i]); x1 ^= x0; }
    x0 += k1;  x1 += ks2 + 4u;
    #pragma unroll
    for (int i = 0; i < 4; ++i) { x0 += x1; x1 = rotl_(x1, RA[i]); x1 ^= x0; }
    x0 += ks2; x1 += k0 + 5u;
    o0 = x0; o1 = x1;
}

// ---------------------------------------------------------------------------
// Kernel 4: per-superpixel attention. One wave per superpixel, 2 waves/block.
// WMMA f32 for qk^T (32x32x32) and attn*v (32x64x32); atomicAdd scatter.
// ---------------------------------------------------------------------------
__global__ __launch_bounds__(64) void k_attn(
    const float* __restrict__ q, const float* __restrict__ kmat,
    const float* __restrict__ v, const int* __restrict__ labels,
    const float* __restrict__ topv, const int* __restrict__ topi,
    float* __restrict__ out)
{
    __shared__ float qs[2][32 * 33];
    __shared__ float ks[2][32 * 33];
    __shared__ float vs[2][32 * 65];
    __shared__ float at[2][32 * 33];
    __shared__ float pl[2][32], ll[2][32], simsl[2][32];
    __shared__ int   idxl[2][32], smp[2][NSAMP];
    __shared__ float psmp[2][NSAMP];

    const int tid  = threadIdx.x;
    const int lane = tid & 31;
    const int w    = tid >> 5;
    const int kidx = blockIdx.x * 2 + w;
    const int t    = lane;

    // --- gather indices, sims, p, logits, masked q/k, masked+weighted v ---
    int   myidx = topi[kidx * 32 + t];
    float mysim = topv[kidx * 32 + t];
    float ssum  = mysim;
    #pragma unroll
    for (int o = 16; o > 0; o >>= 1) ssum += __shfl_xor(ssum, o, 32);
    float p = mysim / ssum;
    pl[w][t]    = p;
    ll[w][t]    = logf(p + 1e-12f);
    simsl[w][t] = mysim;
    idxl[w][t]  = myidx;
    float m = (labels[myidx] == kidx) ? 1.f : 0.f;
    for (int c = 0; c < 32; ++c) {
        qs[w][t * 33 + c] = m * q   [(size_t)c * NPIX + myidx];
        ks[w][t * 33 + c] = m * kmat[(size_t)c * NPIX + myidx];
    }
    float wsc = m * mysim;                  // mask * sims (weight on v_sp)
    for (int c = 0; c < 64; ++c)
        vs[w][t * 65 + c] = wsc * v[(size_t)c * NPIX + myidx];
    __syncthreads();

    // --- attn = exp(scale * q_sp * k_sp^T) via f32 WMMA ---
    const int   mrow  = lane & 15;
    const int   kh    = (lane < 16) ? 0 : 2;
    const float scale = 0.17677669529663687f;   // 32^-0.5
    for (int ti = 0; ti < 2; ++ti)
        for (int si = 0; si < 2; ++si) {
            v8f acc = {0.f, 0.f, 0.f, 0.f, 0.f, 0.f, 0.f, 0.f};
            #pragma unroll
            for (int kq = 0; kq < 8; ++kq) {
                const int c0 = kq * 4 + kh;
                v2f a, b;
                a.x = qs[w][(ti * 16 + mrow) * 33 + c0];
                a.y = qs[w][(ti * 16 + mrow) * 33 + c0 + 1];
                b.x = ks[w][(si * 16 + mrow) * 33 + c0];
                b.y = ks[w][(si * 16 + mrow) * 33 + c0 + 1];
                acc = __builtin_amdgcn_wmma_f32_16x16x4_f32(
                        false, a, false, b, (short)0, acc, false, false);
            }
            const int rb = ti * 16 + ((lane < 16) ? 0 : 8);
            #pragma unroll
            for (int r = 0; r < 8; ++r)
                at[w][(rb + r) * 33 + si * 16 + mrow] = expf(acc[r] * scale);
        }
    __syncthreads();

    // --- JAX-exact categorical sampling (threefry gumbel, key=(0,123)) ---
    if (lane < NSAMP) {
        const unsigned HALF = 276480u;           // 576*30*32 / 2
        const unsigned base = (unsigned)(kidx * NSAMP + lane) * 32u;
        float best = -3.4e38f; int bi = 0;
        for (int tt = 0; tt < 32; ++tt) {
            unsigned i = base + (unsigned)tt;
            unsigned j = (i < HALF) ? i : i - HALF;
            unsigned r0, r1;
            threefry2x32(0u, 123u, j, j + HALF, r0, r1);
            unsigned bits = (i < HALF) ? r0 : r1;
            float f = __uint_as_float(0x3f800000u | (bits >> 9)) - 1.0f;
            const float TINY = 1.1754943508222875e-38f;
            float u = f * (1.0f - TINY) + TINY;
            u = fmaxf(u, TINY);
            float g = -logf(-logf(u));
            float val = ll[w][tt] + g;
            if (val > best) { best = val; bi = tt; }
        }
        smp[w][lane]  = bi;
        psmp[w][lane] = pl[w][bi];
    }
    __syncthreads();

    // --- est / normz, scale attn rows ---
    {
        float e = 0.f;
        for (int s = 0; s < NSAMP; ++s)
            e += at[w][t * 33 + smp[w][s]] / psmp[w][s];
        e *= (1.f / (float)NSAMP);
        float nz = 1.0f / e;
        for (int s = 0; s < 32; ++s) at[w][t * 33 + s] *= nz;
    }
    __syncthreads();

    // --- out = attn * v_sp  (f32 WMMA), then *sims[t], scatter-add ---
    for (int ti = 0; ti < 2; ++ti)
        for (int ci = 0; ci < 4; ++ci) {
            v8f acc = {0.f, 0.f, 0.f, 0.f, 0.f, 0.f, 0.f, 0.f};
            #pragma unroll
            for (int kq = 0; kq < 8; ++kq) {
                const int s0 = kq * 4 + kh;
                v2f a, b;
                a.x = at[w][(ti * 16 + mrow) * 33 + s0];
                a.y = at[w][(ti * 16 + mrow) * 33 + s0 + 1];
                b.x = vs[w][s0 * 65 + ci * 16 + mrow];
                b.y = vs[w][(s0 + 1) * 65 + ci * 16 + mrow];
                acc = __builtin_amdgcn_wmma_f32_16x16x4_f32(
                        false, a, false, b, (short)0, acc, false, false);
            }
            const int rb  = ti * 16 + ((lane < 16) ? 0 : 8);
            const int col = ci * 16 + mrow;
            #pragma unroll
            for (int r = 0; r < 8; ++r) {
                const int row = rb + r;
                float val = acc[r] * simsl[w][row];
                atomicAdd(&out[(size_t)col * NPIX + idxl[w][row]], val);
            }
        }
}

// ---------------------------------------------------------------------------
extern "C" void kernel_launch(void* const* d_in, const int* in_sizes, int n_in,
                              void* d_out, int out_size, void* d_ws, size_t ws_size,
                              hipStream_t stream)
{
    (void)in_sizes; (void)n_in; (void)out_size; (void)ws_size;
    const float* x     = (const float*)d_in[0];
    const float* aff   = (const float*)d_in[1];
    const float* Wq    = (const float*)d_in[2];
    const float* Wk    = (const float*)d_in[3];
    const float* Wv    = (const float*)d_in[4];
    const float* gamma = (const float*)d_in[5];
    const float* beta  = (const float*)d_in[6];
    float* out = (float*)d_out;

    char* ws = (char*)d_ws;
    size_t off = 0;
    float* q      = (float*)(ws + off); off += (size_t)QKD  * NPIX * 4;
    float* k      = (float*)(ws + off); off += (size_t)QKD  * NPIX * 4;
    float* v      = (float*)(ws + off); off += (size_t)CDIM * NPIX * 4;
    int*   labels = (int*)  (ws + off); off += (size_t)NPIX * 4;
    float* topv   = (float*)(ws + off); off += (size_t)NSP * TOPK_ * 4;
    int*   topi   = (int*)  (ws + off); off += (size_t)NSP * TOPK_ * 4;

    k_ln_qkv <<<NPIX / 64, 64, 0, stream>>>(x, Wq, Wk, Wv, gamma, beta, q, k, v, out);
    k_labels <<<NPIX / 256, 256, 0, stream>>>(aff, labels);
    k_topk   <<<NSP, 128, 0, stream>>>(aff, topv, topi);
    k_attn   <<<NSP / 2, 64, 0, stream>>>(q, k, v, labels, topv, topi, out);
}